// GNN_17386027614800
// MI455X (gfx1250) — compile-verified
//
#include <hip/hip_runtime.h>

#define NN 10000  // N_NODES

typedef __attribute__((ext_vector_type(2))) float v2f;
typedef __attribute__((ext_vector_type(8))) float v8f;

// ---------------- degree count ----------------
__global__ void sage_count(const int* __restrict__ dst, float* __restrict__ cnt, int E) {
  int e = blockIdx.x * blockDim.x + threadIdx.x;
  if (e < E) atomicAdd(&cnt[dst[e]], 1.0f);
}

__global__ void sage_invcnt(const float* __restrict__ cnt, float* __restrict__ invc, int n) {
  int i = blockIdx.x * blockDim.x + threadIdx.x;
  if (i < n) invc[i] = 1.0f / fmaxf(cnt[i], 1.0f);
}

// ---------------- edge scatter-add (gather x[src], atomic-add into agg[dst]) ----------
// one thread per (edge, 4-feature chunk); x rows are 16B aligned for float4
__global__ void sage_scatter(const int* __restrict__ src, const int* __restrict__ dst,
                             const float* __restrict__ x, float* __restrict__ agg,
                             int E, int C) {
  const int chunks = C >> 2;
  long long t = (long long)blockIdx.x * blockDim.x + threadIdx.x;
  if (t >= (long long)E * chunks) return;
  int e = (int)(t / chunks);
  int c = (int)(t % chunks) << 2;
  int s = src[e], d = dst[e];
  float4 v = *(const float4*)(x + (size_t)s * C + c);
  float* p = agg + (size_t)d * C + c;
  atomicAdd(p + 0, v.x);
  atomicAdd(p + 1, v.y);
  atomicAdd(p + 2, v.z);
  atomicAdd(p + 3, v.w);
}

// ---------------- fused SAGE layer GEMM via V_WMMA_F32_16X16X4_F32 ----------------
// out[M,N] = act( (agg ⊙ invc) @ Wl  +  xin @ Wr  +  bias )
// One wave computes one 16x16 output tile; K looped in steps of 4 (fp32 matrix core).
template <int RELU>
__global__ void sage_gemm(const float* __restrict__ agg, const float* __restrict__ invc,
                          const float* __restrict__ xin,
                          const float* __restrict__ Wl, const float* __restrict__ Wr,
                          const float* __restrict__ bias, float* __restrict__ out,
                          int M, int K, int N) {
  const int lane = threadIdx.x & 31;
  const int wave = threadIdx.x >> 5;
  const int wavesPerBlock = blockDim.x >> 5;
  const int nTilesN = N >> 4;
  const int nTiles = (M >> 4) * nTilesN;
  const int tile = blockIdx.x * wavesPerBlock + wave;   // wave-uniform
  if (tile >= nTiles) return;

  const int m0 = (tile / nTilesN) << 4;
  const int n0 = (tile % nTilesN) << 4;

  const int rA   = lane & 15;    // A row / B column within tile
  const int hi   = lane >> 4;    // lane half
  const int kSel = hi << 1;      // lanes 0-15 hold K=k,k+1 ; lanes 16-31 hold K=k+2,k+3
  const int colB = n0 + rA;

  const float ic = invc[m0 + rA];
  const float* aggRow = agg + (size_t)(m0 + rA) * K;
  const float* xRow   = xin + (size_t)(m0 + rA) * K;

  v8f acc = {};
  for (int k = 0; k < K; k += 4) {
    const int kr = k + kSel;
    v2f al, ar, bl, br;
    // A operands (16x4 f32, ISA layout): mean folded in via invc
    al.x = aggRow[kr]     * ic;
    al.y = aggRow[kr + 1] * ic;
    ar.x = xRow[kr];
    ar.y = xRow[kr + 1];
    // B operands (4x16 f32): VGPR0 -> rows k/k+2, VGPR1 -> rows k+1/k+3
    bl.x = Wl[(size_t)kr * N + colB];
    bl.y = Wl[(size_t)(kr + 1) * N + colB];
    br.x = Wr[(size_t)kr * N + colB];
    br.y = Wr[(size_t)(kr + 1) * N + colB];

    acc = __builtin_amdgcn_wmma_f32_16x16x4_f32(false, al, false, bl,
                                                (short)0, acc, false, false);
    acc = __builtin_amdgcn_wmma_f32_16x16x4_f32(false, ar, false, br,
                                                (short)0, acc, false, false);
  }

  // C/D layout: VGPR v, lanes 0-15 -> M=v, lanes 16-31 -> M=v+8; N = n0 + (lane&15)
  const float b = bias[colB];
#pragma unroll
  for (int v = 0; v < 8; ++v) {
    float r = acc[v] + b;
    if (RELU) r = fmaxf(r, 0.0f);
    out[(size_t)(m0 + v + (hi << 3)) * N + colB] = r;
  }
}

extern "C" void kernel_launch(void* const* d_in, const int* in_sizes, int n_in,
                              void* d_out, int out_size, void* d_ws, size_t ws_size,
                              hipStream_t stream) {
  const float* x   = (const float*)d_in[0];
  const int*  eidx = (const int*)d_in[1];   // [2,E] flattened (src row, dst row)
  const float* W1l = (const float*)d_in[2];
  const float* W1r = (const float*)d_in[3];
  const float* b1  = (const float*)d_in[4];
  const float* W2l = (const float*)d_in[5];
  const float* W2r = (const float*)d_in[6];
  const float* b2  = (const float*)d_in[7];

  const int E = in_sizes[1] / 2;
  const int N = NN;
  const int* src = eidx;
  const int* dst = eidx + E;

  // workspace layout (floats): [agg1 N*128][agg2 N*256][cnt N][invc N][h N*256]
  float* ws   = (float*)d_ws;
  float* agg1 = ws;
  float* agg2 = agg1 + (size_t)N * 128;
  float* cnt  = agg2 + (size_t)N * 256;
  float* invc = cnt + N;
  float* h    = invc + N;

  // zero the accumulators + counts (contiguous region), capture-safe
  hipMemsetAsync(agg1, 0, ((size_t)N * 128 + (size_t)N * 256 + N) * sizeof(float), stream);

  const int tb = 256;
  sage_count<<<(E + tb - 1) / tb, tb, 0, stream>>>(dst, cnt, E);
  sage_invcnt<<<(N + tb - 1) / tb, tb, 0, stream>>>(cnt, invc, N);

  // layer 1: aggregate x (C=128), fused GEMM -> h (relu)
  long long w1 = (long long)E * (128 / 4);
  sage_scatter<<<(int)((w1 + tb - 1) / tb), tb, 0, stream>>>(src, dst, x, agg1, E, 128);
  int tiles1 = (N / 16) * (256 / 16);
  sage_gemm<1><<<(tiles1 + 7) / 8, 256, 0, stream>>>(agg1, invc, x, W1l, W1r, b1, h,
                                                     N, 128, 256);

  // layer 2: aggregate h (C=256), fused GEMM -> d_out (no relu)
  long long w2 = (long long)E * (256 / 4);
  sage_scatter<<<(int)((w2 + tb - 1) / tb), tb, 0, stream>>>(src, dst, h, agg2, E, 256);
  int tiles2 = (N / 16) * (128 / 16);
  sage_gemm<0><<<(tiles2 + 7) / 8, 256, 0, stream>>>(agg2, invc, h, W2l, W2r, b2,
                                                     (float*)d_out, N, 256, 128);
}